// DecoderAttention_8744553414961
// MI455X (gfx1250) — compile-verified
//
#include <hip/hip_runtime.h>

typedef __attribute__((ext_vector_type(2))) float v2f;
typedef __attribute__((ext_vector_type(8))) float v8f;

namespace {
constexpr int kB     = 16;    // batch
constexpr int kH     = 16;    // heads
constexpr int kD     = 128;   // head dim
constexpr int kBS    = 16;    // cache block size
constexpr int kMAXC  = 1024;  // max context
constexpr int kNTBL  = 96;    // block_tables width
constexpr int kPT    = 32;    // prompt-table prefix (512/16)
constexpr int kPARTS = 4;     // split-K partitions per (b,h)
constexpr int kKPP   = kMAXC / kPARTS;  // 256 keys per partition
constexpr int kBPP   = kKPP / kBS;      // 16 blocks per partition
constexpr float kSCALE  = 0.08838834764831845f;
constexpr float kNEGINF = -1e30f;
}

// ---------------------------------------------------------------------------
// Kernel 1: one workgroup per (b, h, partition). Computes partition-local
// softmax partials: m (max), l (sum of exp), o[d] (unnormalized sum p*V).
// ---------------------------------------------------------------------------
__global__ __launch_bounds__(256) void pa_part_kernel(
    const float* __restrict__ query,
    const float* __restrict__ k_new,
    const float* __restrict__ v_new,
    const float* __restrict__ key_cache,    // [NBLKS, H, 16, 128]
    const float* __restrict__ value_cache,  // [NBLKS, H, 128, 16]
    const int*   __restrict__ block_tables, // [B, 96]
    const int*   __restrict__ context_lens, // [B]
    const float* __restrict__ attn_bias,    // [B, H, 1, 1024]
    float*       __restrict__ ws_o,         // [B*H*PARTS, 128]
    float*       __restrict__ ws_m,         // [B*H*PARTS]
    float*       __restrict__ ws_l)         // [B*H*PARTS]
{
    __shared__ float q_lds[kD];
    __shared__ float s_lds[kKPP];    // scores, then probabilities (in place)
    __shared__ float red[256];
    __shared__ float out_lds[kD];

    const int tid  = threadIdx.x;
    const int lane = tid & 31;
    const int wave = tid >> 5;       // 0..7
    const int half = lane >> 4;      // 0 or 1 (lane group)
    const int m    = lane & 15;      // row index within tile

    const int pidx = blockIdx.x;          // bh * kPARTS + part
    const int bh   = pidx >> 2;
    const int part = pidx & (kPARTS - 1);
    const int b    = bh >> 4;
    const int h    = bh & 15;

    const int ctx   = context_lens[b];
    const int ctxm1 = ctx - 1;
    const int kbeg  = part * kKPP;

    // Dead partition: entire context ends before us. Workgroup-uniform branch.
    if (kbeg >= ctx) {
        if (tid < kD) ws_o[(size_t)pidx * kD + tid] = 0.0f;
        if (tid == 0) { ws_m[pidx] = kNEGINF; ws_l[pidx] = 0.0f; }
        return;
    }

    if (tid < kD) {
        q_lds[tid]   = query[((size_t)b * kH + h) * kD + tid];
        out_lds[tid] = 0.0f;
    }
    __syncthreads();

    const float* biasrow = attn_bias + ((size_t)b * kH + h) * kMAXC;
    const int jbase = part * kBPP + wave * 2;   // 2 blocks per wave

    // ---------------- QK: scores via V_WMMA_F32_16X16X4_F32 ----------------
    for (int jj = 0; jj < 2; ++jj) {
        const int j  = jbase + jj;
        const int k0 = j * kBS;
        if (k0 >= ctx) {                        // wave-uniform tail skip
            if (lane < kBS) s_lds[k0 - kbeg + lane] = kNEGINF;
            continue;
        }
        const int blk = block_tables[b * kNTBL + kPT + j];
        const int kg  = k0 + m;
        // virtual KV-cache update: key row ctx-1 comes from k_new
        const float* krow = (kg == ctxm1)
            ? (k_new + ((size_t)b * kH + h) * kD)
            : (key_cache + (((size_t)blk * kH + h) * kBS + m) * kD);

        v8f acc0 = {0.f,0.f,0.f,0.f,0.f,0.f,0.f,0.f};
        v8f acc1 = {0.f,0.f,0.f,0.f,0.f,0.f,0.f,0.f};
        #pragma unroll
        for (int c = 0; c < 32; c += 2) {
            // A: 16x4 f32 K-tile; lane l<16 -> K[l,4c+0..1], lane 16+l -> K[l,4c+2..3]
            v2f a0 = *(const v2f*)(krow + 4 * c + 2 * half);
            v2f b0;                              // B: q chunk broadcast over 16 cols
            b0.x = q_lds[4 * c + half];
            b0.y = q_lds[4 * c + 2 + half];
            acc0 = __builtin_amdgcn_wmma_f32_16x16x4_f32(false, a0, false, b0,
                                                         (short)0, acc0, false, false);
            v2f a1 = *(const v2f*)(krow + 4 * (c + 1) + 2 * half);
            v2f b1;
            b1.x = q_lds[4 * (c + 1) + half];
            b1.y = q_lds[4 * (c + 1) + 2 + half];
            acc1 = __builtin_amdgcn_wmma_f32_16x16x4_f32(false, a1, false, b1,
                                                         (short)0, acc1, false, false);
        }
        // C layout: VGPR v holds key (v + 8*half); columns replicated
        if (m == 0) {
            #pragma unroll
            for (int v = 0; v < 8; ++v) {
                const int kk = k0 + v + 8 * half;
                float sc = (acc0[v] + acc1[v]) * kSCALE + biasrow[kk];
                if (kk >= ctx) sc = kNEGINF;
                s_lds[kk - kbeg] = sc;
            }
        }
    }
    __syncthreads();

    // ---------------- partition-local softmax partials ----------------
    red[tid] = s_lds[tid];                       // 256 scores, 256 threads
    __syncthreads();
    for (int st = 128; st > 0; st >>= 1) {
        if (tid < st) red[tid] = fmaxf(red[tid], red[tid + st]);
        __syncthreads();
    }
    const float mx = red[0];
    __syncthreads();

    const float p = __expf(s_lds[tid] - mx);
    s_lds[tid] = p;
    red[tid]   = p;
    __syncthreads();
    for (int st = 128; st > 0; st >>= 1) {
        if (tid < st) red[tid] += red[tid + st];
        __syncthreads();
    }
    const float lsum = red[0];
    __syncthreads();

    // ---------------- PV: o += p * V via WMMA ----------------
    v8f oacc[8];
    #pragma unroll
    for (int t = 0; t < 8; ++t) oacc[t] = (v8f){0.f,0.f,0.f,0.f,0.f,0.f,0.f,0.f};

    for (int jj = 0; jj < 2; ++jj) {
        const int j  = jbase + jj;
        const int k0 = j * kBS;
        if (k0 >= ctx) continue;                 // p == 0 there
        const int blk = block_tables[b * kNTBL + kPT + j];
        const float* vbase = value_cache + ((size_t)blk * kH + h) * (kD * kBS);
        #pragma unroll
        for (int kk0 = 0; kk0 < kBS; kk0 += 4) {
            const int kb = k0 + kk0;
            const int g0 = kb + half;            // global key of B rows 0/1
            const int g1 = kb + 2 + half;        // global key of B rows 2/3
            float p0 = s_lds[g0 - kbeg]; if (g0 == ctxm1) p0 = 0.0f;  // v_new later
            float p1 = s_lds[g1 - kbeg]; if (g1 == ctxm1) p1 = 0.0f;
            v2f pb; pb.x = p0; pb.y = p1;
            #pragma unroll
            for (int t = 0; t < 8; ++t) {
                // A: value_cache is [d,16]: lane m -> V[16t+m, kk0+2*half ..+1]
                v2f a = *(const v2f*)(vbase + (t * kBS + m) * kBS + kk0 + 2 * half);
                oacc[t] = __builtin_amdgcn_wmma_f32_16x16x4_f32(false, a, false, pb,
                                                                (short)0, oacc[t],
                                                                false, false);
            }
        }
    }
    // cross-wave reduce: ds_add_f32 on LDS; d = 16t + v + 8*half, lanes 0/16
    if (m == 0) {
        #pragma unroll
        for (int t = 0; t < 8; ++t)
            #pragma unroll
            for (int v = 0; v < 8; ++v)
                atomicAdd(&out_lds[t * kBS + v + 8 * half], oacc[t][v]);
    }
    __syncthreads();

    // ---------------- write partials ----------------
    if (tid < kD) {
        float o = out_lds[tid];
        if (ctxm1 >= kbeg && ctxm1 < kbeg + kKPP)   // freshly-written KV position
            o += s_lds[ctxm1 - kbeg] * v_new[((size_t)b * kH + h) * kD + tid];
        ws_o[(size_t)pidx * kD + tid] = o;
    }
    if (tid == 0) { ws_m[pidx] = mx; ws_l[pidx] = lsum; }
}

// ---------------------------------------------------------------------------
// Kernel 2: combine the kPARTS partials per (b,h) with softmax rescaling.
// ---------------------------------------------------------------------------
__global__ __launch_bounds__(kD) void pa_combine_kernel(
    const float* __restrict__ ws_o,
    const float* __restrict__ ws_m,
    const float* __restrict__ ws_l,
    float*       __restrict__ out)          // [B, 1, H*128]
{
    const int bh  = blockIdx.x;
    const int tid = threadIdx.x;

    float mi[kPARTS];
    float M = kNEGINF;
    #pragma unroll
    for (int i = 0; i < kPARTS; ++i) {
        mi[i] = ws_m[bh * kPARTS + i];
        M = fmaxf(M, mi[i]);
    }
    float L = 0.0f, acc = 0.0f;
    #pragma unroll
    for (int i = 0; i < kPARTS; ++i) {
        const float w = __expf(mi[i] - M);      // dead partitions: w -> 0, l == 0
        L   += ws_l[bh * kPARTS + i] * w;
        acc += ws_o[(size_t)(bh * kPARTS + i) * kD + tid] * w;
    }
    out[(size_t)bh * kD + tid] = acc / L;
}

extern "C" void kernel_launch(void* const* d_in, const int* in_sizes, int n_in,
                              void* d_out, int out_size, void* d_ws, size_t ws_size,
                              hipStream_t stream) {
    (void)in_sizes; (void)n_in; (void)out_size; (void)ws_size;
    const float* query       = (const float*)d_in[0];
    const float* k_new       = (const float*)d_in[1];
    const float* v_new       = (const float*)d_in[2];
    const float* key_cache   = (const float*)d_in[3];
    const float* value_cache = (const float*)d_in[4];
    // d_in[5] = slot_mapping (implied by context_lens; unused)
    const int*   block_tables = (const int*)d_in[6];
    const int*   context_lens = (const int*)d_in[7];
    const float* attn_bias    = (const float*)d_in[8];
    // d_in[9] = max_prompt_len (constant 512 -> kPT = 32)
    float* out = (float*)d_out;

    constexpr int nPart = kB * kH * kPARTS;               // 1024
    float* ws_o = (float*)d_ws;                           // nPart * 128
    float* ws_m = ws_o + (size_t)nPart * kD;              // nPart
    float* ws_l = ws_m + nPart;                           // nPart

    pa_part_kernel<<<dim3(nPart), dim3(256), 0, stream>>>(
        query, k_new, v_new, key_cache, value_cache,
        block_tables, context_lens, attn_bias, ws_o, ws_m, ws_l);

    pa_combine_kernel<<<dim3(kB * kH), dim3(kD), 0, stream>>>(ws_o, ws_m, ws_l, out);
}